// PINN_16320875725160
// MI455X (gfx1250) — compile-verified
//
#include <hip/hip_runtime.h>
#include <hip/hip_bf16.h>
#include <math.h>

// CDNA5 / gfx1250 PINN beam kernel.
// Layer-2 jet propagation (4 orders of 16x16 @ 16x16) runs on the matrix pipe
// via V_WMMA_F32_16X16X4_F32 (f32 kept for 3rd-derivative accuracy).
// Each wave32 processes a tile of 16 collocation points:
//   - lane l: point n = l&15, half = l>>4
//   - lane computes tanh-jets of its 8 layer-1 hidden units (k = 4c+2*half+{0,1})
//     which is exactly the B-matrix fragment layout for the 4 chained K=4 WMMAs,
//     and matches the A-matrix (W2) fragment k-indices for the same lane.
//   - D fragment: lane holds z2 jets for hidden rows m2 = 8*half + r.
//   - 2x16 output layer: per-lane FMAs + __shfl_xor(.,16) cross-half reduce.
// tanh is branch-free via v_exp_f32 + v_rcp_f32 (+1 NR step) so the hot loop
// has no EXEC-mask churn; stores are split across halves (3 per lane) so EXEC
// stays all-ones for the entire loop.

typedef __attribute__((ext_vector_type(2))) float v2f;
typedef __attribute__((ext_vector_type(8))) float v8f;

#define PINN_L   2.0f
#define PINN_EA  1000.0f
#define PINN_EI  100.0f

__device__ __forceinline__ v8f wmma_f32_16x16x4(v2f a, v2f b, v8f c) {
    return __builtin_amdgcn_wmma_f32_16x16x4_f32(
        /*neg_a=*/false, a, /*neg_b=*/false, b,
        /*c_mod=*/(short)0, c, /*reuse_a=*/false, /*reuse_b=*/false);
}

// Branch-free tanh: tanh(z) = 1 - 2/(exp(2z)+1).
// v_exp_f32 + v_rcp_f32 with one Newton-Raphson refinement of the reciprocal.
// exp2 saturating to 0 gives exactly t = -1; at +inf the NR step yields NaN (inf*0).
__device__ __forceinline__ float tanh_fast(float z) {
    const float LOG2E_X2 = 2.885390081777926814f;   // 2*log2(e)
    float e  = __builtin_amdgcn_exp2f(z * LOG2E_X2);
    float d  = e + 1.0f;
    float r  = __builtin_amdgcn_rcpf(d);
    r = fmaf(fmaf(-d, r, 1.0f), r, r);              // NR: r += r*(1 - d*r)
    return fmaf(-2.0f, r, 1.0f);
}

__global__ __launch_bounds__(256) void pinn_beam_kernel(
    const float* __restrict__ x,
    const float* __restrict__ W1, const float* __restrict__ b1,
    const float* __restrict__ W2, const float* __restrict__ b2,
    const float* __restrict__ W3, const float* __restrict__ b3,
    float* __restrict__ out, int npts)
{
    const int lane  = threadIdx.x & 31;
    const int n     = lane & 15;     // point index within tile / A row m / D col n
    const int half  = lane >> 4;     // k-group select

    // ---- hoisted constant weights (per-lane fragments) ----
    float w1c[8], b1v[8];            // layer-1: c = W1/L (dz1/dx), bias
    v2f   aW2[4];                    // A-matrix fragments of W2 for the 4 K-chunks
    #pragma unroll
    for (int c = 0; c < 4; ++c) {
        const int k0 = 4 * c + 2 * half;
        w1c[2*c]   = W1[k0]     * (1.0f / PINN_L);
        w1c[2*c+1] = W1[k0 + 1] * (1.0f / PINN_L);
        b1v[2*c]   = b1[k0];
        b1v[2*c+1] = b1[k0 + 1];
        aW2[c].x   = W2[n * 16 + k0];
        aW2[c].y   = W2[n * 16 + k0 + 1];
    }
    v8f cbias;                       // b2 broadcast in C/D layout (order-0 init)
    float w3u[8], w3w[8];            // W3 rows for the half's 8 hidden units
    #pragma unroll
    for (int r = 0; r < 8; ++r) {
        cbias[r] = b2[8 * half + r];
        w3u[r]   = W3[ 0 + 8 * half + r];
        w3w[r]   = W3[16 + 8 * half + r];
    }
    const float b3u = b3[0], b3w = b3[1];

    const int waves_total = (int)(gridDim.x * (blockDim.x >> 5));
    const int wave_id     = (int)(blockIdx.x * (blockDim.x >> 5) + (threadIdx.x >> 5));
    const int ntiles      = npts >> 4;

    // Output base for this lane's half: half 0 stores (u, w, wx),
    // half 1 stores (N_ax, M_bd, Q_sh). Keeps EXEC all-ones everywhere.
    const int obase = half * 3 * npts;

    for (int t = wave_id; t < ntiles; t += waves_total) {
        const int p  = t * 16 + n;
        const float xr = x[p];

        // ---- layer-1 tanh jets for this lane's 8 hidden units ----
        float g0[8], g1[8], g2[8], g3[8];
        #pragma unroll
        for (int u = 0; u < 8; ++u) {
            const float c1  = w1c[u];
            const float tt  = tanh_fast(fmaf(c1, xr, b1v[u]));
            const float ss  = fmaf(-tt, tt, 1.0f);
            const float c1s = c1 * c1;
            g0[u] = tt;
            g1[u] = ss * c1;
            g2[u] = -2.0f * tt * ss * c1s;
            g3[u] = -2.0f * c1s * c1 * ss * fmaf(-2.0f * tt, tt, ss);
        }

        // ---- layer-2: z2 jets = W2 @ a1 jets (16 chained f32 WMMAs) ----
        v8f z0 = cbias;              // order 0 gets bias b2
        v8f z1v = (v8f)0, z2v = (v8f)0, z3v = (v8f)0;
        #pragma unroll
        for (int c = 0; c < 4; ++c) {
            v2f bb;
            bb.x = g0[2*c]; bb.y = g0[2*c+1];  z0  = wmma_f32_16x16x4(aW2[c], bb, z0);
            bb.x = g1[2*c]; bb.y = g1[2*c+1];  z1v = wmma_f32_16x16x4(aW2[c], bb, z1v);
            bb.x = g2[2*c]; bb.y = g2[2*c+1];  z2v = wmma_f32_16x16x4(aW2[c], bb, z2v);
            bb.x = g3[2*c]; bb.y = g3[2*c+1];  z3v = wmma_f32_16x16x4(aW2[c], bb, z3v);
        }

        // ---- layer-2 tanh jet composition + layer-3 partial matvec ----
        float pu0 = 0.f, pu1 = 0.f, pw0 = 0.f, pw1 = 0.f, pw2 = 0.f, pw3 = 0.f;
        #pragma unroll
        for (int r = 0; r < 8; ++r) {
            const float t2 = tanh_fast(z0[r]);
            const float s2 = fmaf(-t2, t2, 1.0f);
            const float d1 = z1v[r], d2 = z2v[r], d3 = z3v[r];
            const float a0 = t2;
            const float a1 = s2 * d1;
            const float a2 = s2 * d2 - 2.0f * t2 * s2 * d1 * d1;
            const float a3 = s2 * d3 - 6.0f * t2 * s2 * d1 * d2
                           - 2.0f * s2 * fmaf(-2.0f * t2, t2, s2) * d1 * d1 * d1;
            pu0 = fmaf(w3u[r], a0, pu0);
            pu1 = fmaf(w3u[r], a1, pu1);
            pw0 = fmaf(w3w[r], a0, pw0);
            pw1 = fmaf(w3w[r], a1, pw1);
            pw2 = fmaf(w3w[r], a2, pw2);
            pw3 = fmaf(w3w[r], a3, pw3);
        }

        // ---- cross-half reduction (halves hold disjoint hidden-unit ranges) ----
        pu0 += __shfl_xor(pu0, 16, 32);
        pu1 += __shfl_xor(pu1, 16, 32);
        pw0 += __shfl_xor(pw0, 16, 32);
        pw1 += __shfl_xor(pw1, 16, 32);
        pw2 += __shfl_xor(pw2, 16, 32);
        pw3 += __shfl_xor(pw3, 16, 32);

        // ---- beam quantities; every lane stores 3 values (no divergence) ----
        const float u    = pu0 + b3u;
        const float w    = pw0 + b3w;
        const float wx   = pw1;
        const float Nax  = PINN_EA * fmaf(0.5f * wx, wx, pu1);
        const float Mbd  = -PINN_EI * pw2;
        const float Qsh  = fmaf(Nax, wx, -PINN_EI * pw3);

        const float o0 = half ? Nax : u;
        const float o1 = half ? Mbd : w;
        const float o2 = half ? Qsh : wx;
        out[obase + p]             = o0;
        out[obase + npts + p]      = o1;
        out[obase + 2 * npts + p]  = o2;
    }
}

extern "C" void kernel_launch(void* const* d_in, const int* in_sizes, int n_in,
                              void* d_out, int out_size, void* d_ws, size_t ws_size,
                              hipStream_t stream) {
    const float* x  = (const float*)d_in[0];
    const float* W1 = (const float*)d_in[1];
    const float* b1 = (const float*)d_in[2];
    const float* W2 = (const float*)d_in[3];
    const float* b2 = (const float*)d_in[4];
    const float* W3 = (const float*)d_in[5];
    const float* b3 = (const float*)d_in[6];
    float* out = (float*)d_out;
    const int npts = in_sizes[0];   // x is (N,1) -> N floats

    // 1024 blocks x 8 waves = 8192 waves; 65536 tiles -> 8 tiles/wave.
    dim3 grid(1024), block(256);
    pinn_beam_kernel<<<grid, block, 0, stream>>>(x, W1, b1, W2, b2, W3, b3, out, npts);
}